// NerfRender_19000935317749
// MI455X (gfx1250) — compile-verified
//
#include <hip/hip_runtime.h>

typedef __attribute__((ext_vector_type(2))) float v2f;
typedef __attribute__((ext_vector_type(8))) float v8f;

#define WAVES_PER_BLOCK 4
#define TILE_RAYS 16
#define NSAMP 128
#define ROW_STRIDE 132                       // 128 data + 4 pad floats (bank skew + t[128] slot)
#define LDS_PER_ARRAY (TILE_RAYS * ROW_STRIDE)

__global__ __launch_bounds__(WAVES_PER_BLOCK * 32)
void nerf_render_kernel(const float* __restrict__ t_in,
                        const float* __restrict__ sigma_in,
                        const float* __restrict__ color_in,
                        float* __restrict__ out_rgb,
                        float* __restrict__ out_depth,
                        float* __restrict__ out_wi,
                        float* __restrict__ out_t)
{
    __shared__ float smem[WAVES_PER_BLOCK * 2 * LDS_PER_ARRAY];

    const int lane = threadIdx.x & 31;
    const int wave = threadIdx.x >> 5;
    const int tile = blockIdx.x * WAVES_PER_BLOCK + wave;
    const int ray_base = tile * TILE_RAYS;

    float* lt = smem + wave * (2 * LDS_PER_ARRAY);  // t tile  [16][132]
    float* ls = lt + LDS_PER_ARRAY;                 // sigma   [16][132]

    // ---- async stage: 16 rows x 512B each of t and sigma into padded LDS rows ----
    {
        const float* gT = t_in     + (size_t)ray_base * NSAMP;
        const float* gS = sigma_in + (size_t)ray_base * NSAMP;
        const char*  base = (const char*)smem;
        #pragma unroll
        for (int r = 0; r < TILE_RAYS; ++r) {
            unsigned offT = (unsigned)((const char*)(lt + r * ROW_STRIDE) - base) + (unsigned)lane * 16u;
            unsigned offS = (unsigned)((const char*)(ls + r * ROW_STRIDE) - base) + (unsigned)lane * 16u;
            const char* gaT = (const char*)(gT + (size_t)r * NSAMP) + lane * 16;
            const char* gaS = (const char*)(gS + (size_t)r * NSAMP) + lane * 16;
            asm volatile("global_load_async_to_lds_b128 %0, %1, off"
                         :: "v"(offT), "v"(gaT) : "memory");
            asm volatile("global_load_async_to_lds_b128 %0, %1, off"
                         :: "v"(offS), "v"(gaS) : "memory");
        }
        asm volatile("s_wait_asynccnt 0" ::: "memory");
    }
    // dt[127] = 0: duplicate t[127] into the pad slot t[128]
    if (lane < TILE_RAYS) {
        lt[lane * ROW_STRIDE + NSAMP] = lt[lane * ROW_STRIDE + NSAMP - 1];
    }

    // ---- constant triangular B fragments (guessed B layout: lane->N, lane-half->K+{0,2}, vgpr->K parity)
    const int n     = lane & 15;          // column within 16-wide tile (C layout)
    const int kb    = (lane >> 4) << 1;   // K offset for this half-wave
    const int mhalf = (lane >> 4) * 8;    // row offset for C layout
    v2f Bi[4], Be[4];
    #pragma unroll
    for (int s = 0; s < 4; ++s) {
        int k0 = 4 * s + kb, k1 = k0 + 1;
        v2f bi; bi.x = (k0 <= n) ? 1.0f : 0.0f; bi.y = (k1 <= n) ? 1.0f : 0.0f;
        v2f be; be.x = (k0 <  n) ? 1.0f : 0.0f; be.y = (k1 <  n) ? 1.0f : 0.0f;
        Bi[s] = bi; Be[s] = be;
    }

    float off8[8] = {0.f,0.f,0.f,0.f,0.f,0.f,0.f,0.f};  // running cumsum carry per ray-row
    float rAcc[8][3] = {};
    float dAcc[8]    = {};

    for (int c = 0; c < 8; ++c) {
        // ---- chunk-local inclusive & exclusive cumsums via WMMA 16x16x4 f32 ----
        v8f incl = {};
        v8f excl = {};
        #pragma unroll
        for (int s = 0; s < 4; ++s) {
            int k0 = c * 16 + s * 4 + kb;
            const float* tp = lt + (lane & 15) * ROW_STRIDE + k0;
            const float* sp = ls + (lane & 15) * ROW_STRIDE + k0;
            float t0 = tp[0], t1 = tp[1], t2 = tp[2];
            v2f A; A.x = sp[0] * (t1 - t0); A.y = sp[1] * (t2 - t1);   // sigma*dt for K, K+1
            incl = __builtin_amdgcn_wmma_f32_16x16x4_f32(false, A, false, Bi[s],
                                                         (short)0, incl, false, false);
            excl = __builtin_amdgcn_wmma_f32_16x16x4_f32(false, A, false, Be[s],
                                                         (short)0, excl, false, false);
        }

        const int col  = c * 16 + n;
        const int bsrc = (lane & 16) | 15;   // lane holding column 15 of this half
        #pragma unroll
        for (int g = 0; g < 8; ++g) {
            float ev = excl[g];
            float iv = incl[g];
            float Ta = __expf(-(off8[g] + ev));   // transmittance before sample
            float Tb = __expf(-(off8[g] + iv));   // transmittance after sample
            float wi = Ta - Tb;                   // == Ti * (1 - exp(-sdt))
            int   m  = g + mhalf;
            float tc = lt[m * ROW_STRIDE + col];
            size_t ray = (size_t)(ray_base + m);
            const float* cp = color_in + (ray * NSAMP + col) * 3;
            rAcc[g][0] += wi * cp[0];
            rAcc[g][1] += wi * cp[1];
            rAcc[g][2] += wi * cp[2];
            dAcc[g]    += wi * tc;
            out_wi[ray * NSAMP + col] = wi;
            out_t [ray * NSAMP + col] = tc;
            off8[g] += __shfl(iv, bsrc, 32);      // carry = chunk total (column 15)
        }
    }

    // ---- reduce over the 16 sample-columns (independent per half-wave) ----
    #pragma unroll
    for (int g = 0; g < 8; ++g) {
        float r0 = rAcc[g][0], r1 = rAcc[g][1], r2 = rAcc[g][2], dd = dAcc[g];
        #pragma unroll
        for (int x = 8; x >= 1; x >>= 1) {
            r0 += __shfl_xor(r0, x, 32);
            r1 += __shfl_xor(r1, x, 32);
            r2 += __shfl_xor(r2, x, 32);
            dd += __shfl_xor(dd, x, 32);
        }
        if (n == 0) {
            size_t ray = (size_t)(ray_base + g + mhalf);
            out_rgb[ray * 3 + 0] = r0;
            out_rgb[ray * 3 + 1] = r1;
            out_rgb[ray * 3 + 2] = r2;
            out_depth[ray] = dd;
        }
    }
}

extern "C" void kernel_launch(void* const* d_in, const int* in_sizes, int n_in,
                              void* d_out, int out_size, void* d_ws, size_t ws_size,
                              hipStream_t stream) {
    (void)n_in; (void)out_size; (void)d_ws; (void)ws_size;
    const float* t     = (const float*)d_in[0];
    const float* sigma = (const float*)d_in[1];
    const float* color = (const float*)d_in[2];
    const int N = in_sizes[0] / NSAMP;           // 65536 rays

    float* out       = (float*)d_out;
    float* out_rgb   = out;                      // [N,3]
    float* out_depth = out + (size_t)N * 3;      // [N,1]
    float* out_wi    = out + (size_t)N * 4;      // [N,128]
    float* out_t     = out + (size_t)N * 4 + (size_t)N * NSAMP;  // [N,128]

    const int rays_per_block = TILE_RAYS * WAVES_PER_BLOCK;      // 64
    dim3 grid((N + rays_per_block - 1) / rays_per_block);        // 1024
    dim3 block(WAVES_PER_BLOCK * 32);                            // 128
    nerf_render_kernel<<<grid, block, 0, stream>>>(t, sigma, color,
                                                   out_rgb, out_depth, out_wi, out_t);
}